// BandSplit_44220983280159
// MI455X (gfx1250) — compile-verified
//
#include <hip/hip_runtime.h>

typedef __attribute__((ext_vector_type(2))) float v2f;
typedef __attribute__((ext_vector_type(8))) float v8f;

#define N_BANDS 62
#define MAXF    54
#define PDIM    216        // 4 * MAXF
#define D_MODEL 384
#define T_LEN   512
#define F_BINS  1025
#define KSTEPS  54         // PDIM / 4
#define ROWS    64         // bt rows per workgroup
#define LDSTR   220        // padded LDS row stride (words); 220*r mod 64 is a 16-perm

// Band tables replicated from the reference _compute_bands():
// widths: 2 x23, 4 x11, 12 x7, 24 x8, 48 x7, 54 x5, 53 x1  (sum = 1025 bins, 62 bands)
__constant__ int BAND_L[N_BANDS] = {
    0,2,4,6,8,10,12,14,16,18,20,22,24,26,28,30,32,34,36,38,40,42,44,
    46,50,54,58,62,66,70,74,78,82,86,
    90,102,114,126,138,150,162,
    174,198,222,246,270,294,318,342,
    366,414,462,510,558,606,654,
    702,756,810,864,918,
    972
};
__constant__ int BAND_FN[N_BANDS] = {
    2,2,2,2,2,2,2,2,2,2,2,2,2,2,2,2,2,2,2,2,2,2,2,
    4,4,4,4,4,4,4,4,4,4,4,
    12,12,12,12,12,12,12,
    24,24,24,24,24,24,24,24,
    48,48,48,48,48,48,48,
    54,54,54,54,54,
    53
};

__global__ __launch_bounds__(256) void bandsplit_wmma_kernel(
    const float* __restrict__ Xr, const float* __restrict__ Xi,
    const float* __restrict__ nw, const float* __restrict__ W,
    const float* __restrict__ bias, float* __restrict__ out)
{
    __shared__ float Y[ROWS * LDSTR];   // 56,320 B

    const int n      = blockIdx.y;
    const int btBase = blockIdx.x * ROWS;
    const int tid    = threadIdx.x;

    const int   l        = BAND_L[n];
    const int   fn       = BAND_FN[n];
    const float invIndim = 1.0f / (4.0f * (float)fn);

    // ---------------- Phase A: gather + RMSNorm + norm_w -> LDS ----------------
    {
        const int row = tid >> 2;          // 0..63
        const int sub = tid & 3;           // 0..3 (4 lanes per row)
        const int bt  = btBase + row;
        const int b   = bt >> 9;           // T_LEN = 512
        const int t   = bt & (T_LEN - 1);
        const float* nwn = nw + (size_t)n * PDIM;

        float vals[54];
        float sumsq = 0.0f;
        #pragma unroll
        for (int i = 0; i < 54; ++i) {
            const int p   = sub + 4 * i;             // 0..215
            const int c   = p / 108;                 // channel
            const int rem = p - 108 * c;
            const int ri  = rem / MAXF;              // 0=real, 1=imag
            const int f   = rem - MAXF * ri;         // bin within band
            const float* Xp = ri ? Xi : Xr;
            float v = 0.0f;
            if (f < fn) {
                const size_t xi = ((size_t)(b * 2 + c) * T_LEN + t) * F_BINS + (l + f);
                v = Xp[xi];
            }
            vals[i] = v;
            sumsq  += v * v;
        }
        // reduce over the 4 lanes of this row (contiguous lanes within a wave32)
        sumsq += __shfl_xor(sumsq, 1, 32);
        sumsq += __shfl_xor(sumsq, 2, 32);
        const float scale = rsqrtf(sumsq * invIndim + 1e-8f);

        #pragma unroll
        for (int i = 0; i < 54; ++i) {
            const int p = sub + 4 * i;
            Y[row * LDSTR + p] = vals[i] * scale * nwn[p];
        }
    }
    __syncthreads();

    // ---------------- Phase B: WMMA f32 16x16x4 GEMM: Y(64x216) @ W[n](216x384) ----------------
    const int wave = tid >> 5;            // 0..7, each owns 3 column tiles (48 cols)
    const int lane = tid & 31;
    const int half = lane >> 4;           // 0: K=0,1 ; 1: K=2,3 halves
    const int m16  = lane & 15;
    const int colBase = wave * 48;

    v8f acc[4][3];
    #pragma unroll
    for (int mt = 0; mt < 4; ++mt)
        #pragma unroll
        for (int jj = 0; jj < 3; ++jj)
            acc[mt][jj] = (v8f){0.f,0.f,0.f,0.f,0.f,0.f,0.f,0.f};

    // B fragment: VGPR0 -> row 4k + 2*half, VGPR1 -> row 4k + 2*half + 1, col = colBase + jj*16 + m16
    const float* wp = W + (size_t)n * PDIM * D_MODEL
                        + (size_t)(2 * half) * D_MODEL + colBase + m16;

    for (int k = 0; k < KSTEPS; ++k) {
        v2f bfrag[3];
        #pragma unroll
        for (int jj = 0; jj < 3; ++jj) {
            const float b0 = wp[jj * 16];
            const float b1 = wp[jj * 16 + D_MODEL];
            bfrag[jj] = (v2f){b0, b1};
        }
        #pragma unroll
        for (int mt = 0; mt < 4; ++mt) {
            // A fragment: lane m16 reads row (mt*16+m16), cols 4k+2*half .. +1 (8B aligned)
            const v2f afrag = *(const v2f*)&Y[(mt * 16 + m16) * LDSTR + 4 * k + 2 * half];
            #pragma unroll
            for (int jj = 0; jj < 3; ++jj) {
                acc[mt][jj] = __builtin_amdgcn_wmma_f32_16x16x4_f32(
                    /*neg_a=*/false, afrag,
                    /*neg_b=*/false, bfrag[jj],
                    /*c_mod=*/(short)0, acc[mt][jj],
                    /*reuse_a=*/false, /*reuse_b=*/false);
            }
        }
        wp += 4 * D_MODEL;
    }

    // ---------------- Store: C layout VGPR r -> M = r + 8*half, N = m16 ----------------
    #pragma unroll
    for (int jj = 0; jj < 3; ++jj) {
        const int d  = colBase + jj * 16 + m16;
        const float bb = bias[(size_t)n * D_MODEL + d];
        #pragma unroll
        for (int mt = 0; mt < 4; ++mt) {
            #pragma unroll
            for (int r = 0; r < 8; ++r) {
                const int m = mt * 16 + r + 8 * half;
                const size_t o = (((size_t)(btBase + m) * N_BANDS) + n) * D_MODEL + d;
                out[o] = acc[mt][jj][r] + bb;
            }
        }
    }
}

extern "C" void kernel_launch(void* const* d_in, const int* in_sizes, int n_in,
                              void* d_out, int out_size, void* d_ws, size_t ws_size,
                              hipStream_t stream) {
    const float* Xr   = (const float*)d_in[0];   // (B,2,512,1025)
    const float* Xi   = (const float*)d_in[1];   // (B,2,512,1025)
    const float* nw   = (const float*)d_in[2];   // (62,216)
    const float* W    = (const float*)d_in[3];   // (62,216,384)
    const float* bias = (const float*)d_in[4];   // (62,384)
    float* out        = (float*)d_out;           // (B,512,62,384)

    const int B  = in_sizes[0] / (2 * T_LEN * F_BINS);
    const int BT = B * T_LEN;
    dim3 grid(BT / ROWS, N_BANDS);
    bandsplit_wmma_kernel<<<grid, 256, 0, stream>>>(Xr, Xi, nw, W, bias, out);
}